// SpectralConv3d_56710748177059
// MI455X (gfx1250) — compile-verified
//
#include <hip/hip_runtime.h>
#include <math.h>

// ---------------------------------------------------------------------------
// SpectralConv3d for MI455X (gfx1250, wave32, WMMA).
//   x:  [8, 32, 64, 64, 64] f32
//   w1..w4: [32, 32, 16, 16, 16, 2] f32
//   out: [8, 32, 64, 64, 64] f32
// Pipeline: Z-FFT (r2c, keep 32 bins) -> Y-FFT (keep 32) -> X-FFT (keep 16)
//           -> per-frequency complex GEMM via V_WMMA_F32_16X16X4_F32
//           -> inverse X/Y/Z with zero-padding + Hermitian reconstruction.
// ---------------------------------------------------------------------------

struct cpx { float x; float y; };

#define FFT_LINES 128   // threads per FFT block; LDS laid out [elem][line]

// Compile-time twiddles: W64^j = exp(-2*pi*i*j/64) = (COS[j], -SIN[j]).
// With the FFT fully unrolled every index is a constant, so these fold to
// inline float literals (no v_sin/v_cos, no range reduction at runtime).
__device__ constexpr float TW_COS[32] = {
   1.0000000000f,  0.9951847267f,  0.9807852804f,  0.9569403357f,
   0.9238795325f,  0.8819212643f,  0.8314696123f,  0.7730104534f,
   0.7071067812f,  0.6343932842f,  0.5555702330f,  0.4713967368f,
   0.3826834324f,  0.2902846773f,  0.1950903220f,  0.0980171403f,
   0.0000000000f, -0.0980171403f, -0.1950903220f, -0.2902846773f,
  -0.3826834324f, -0.4713967368f, -0.5555702330f, -0.6343932842f,
  -0.7071067812f, -0.7730104534f, -0.8314696123f, -0.8819212643f,
  -0.9238795325f, -0.9569403357f, -0.9807852804f, -0.9951847267f
};
__device__ constexpr float TW_SIN[32] = {
   0.0000000000f,  0.0980171403f,  0.1950903220f,  0.2902846773f,
   0.3826834324f,  0.4713967368f,  0.5555702330f,  0.6343932842f,
   0.7071067812f,  0.7730104534f,  0.8314696123f,  0.8819212643f,
   0.9238795325f,  0.9569403357f,  0.9807852804f,  0.9951847267f,
   1.0000000000f,  0.9951847267f,  0.9807852804f,  0.9569403357f,
   0.9238795325f,  0.8819212643f,  0.8314696123f,  0.7730104534f,
   0.7071067812f,  0.6343932842f,  0.5555702330f,  0.4713967368f,
   0.3826834324f,  0.2902846773f,  0.1950903220f,  0.0980171403f
};

// Forward DIF radix-2 64-pt FFT on one LDS column (stride FFT_LINES).
// Natural-order input, bit-reversed output. Bank-conflict-free: fixed elem
// index across lanes -> consecutive lanes hit consecutive float2 slots.
__device__ __forceinline__ void fft64_col(cpx* c) {
#pragma unroll
  for (int len = 64; len >= 2; len >>= 1) {
    int half = len >> 1;
    int step = 64 / len;           // twiddle index stride into W64 table
#pragma unroll
    for (int base = 0; base < 64; base += len) {
#pragma unroll
      for (int j = 0; j < half; ++j) {
        cpx u = c[(base + j) * FFT_LINES];
        cpx v = c[(base + j + half) * FFT_LINES];
        cpx s; s.x = u.x + v.x; s.y = u.y + v.y;
        cpx d; d.x = u.x - v.x; d.y = u.y - v.y;
        float cs = TW_COS[j * step];
        float sn = -TW_SIN[j * step];
        c[(base + j) * FFT_LINES] = s;
        cpx t; t.x = d.x * cs - d.y * sn; t.y = d.x * sn + d.y * cs;
        c[(base + j + half) * FFT_LINES] = t;
      }
    }
  }
}

__device__ __forceinline__ int brev6(int b) {
  return (int)(__brev((unsigned)b) >> 26);
}

// ---------------- Forward Z: real 64 -> 32 kept complex bins (0..15,17..32)
__global__ __launch_bounds__(FFT_LINES) void k_fft_z_fwd(
    const float* __restrict__ x, cpx* __restrict__ out) {
  __shared__ cpx lds[64 * FFT_LINES];  // 64 KB
  int t = threadIdx.x;
  size_t l = (size_t)blockIdx.x * FFT_LINES + t;  // line = (b,ci,x,y)
  const float4* src = (const float4*)(x + l * 64);
#pragma unroll
  for (int k = 0; k < 16; ++k) {
    float4 v = src[k];
    lds[(4 * k + 0) * FFT_LINES + t] = {v.x, 0.f};
    lds[(4 * k + 1) * FFT_LINES + t] = {v.y, 0.f};
    lds[(4 * k + 2) * FFT_LINES + t] = {v.z, 0.f};
    lds[(4 * k + 3) * FFT_LINES + t] = {v.w, 0.f};
  }
  fft64_col(lds + t);
  cpx* o = out + l * 32;
#pragma unroll
  for (int zk = 0; zk < 32; ++zk) {
    int bin = (zk < 16) ? zk : zk + 1;
    o[zk] = lds[brev6(bin) * FFT_LINES + t];
  }
}

// ---------------- Forward Y: 64 -> 32 kept bins (0..15, 48..63)
__global__ __launch_bounds__(FFT_LINES) void k_fft_y_fwd(
    const cpx* __restrict__ in, cpx* __restrict__ out) {
  __shared__ cpx lds[64 * FFT_LINES];
  int t = threadIdx.x;
  size_t l = (size_t)blockIdx.x * FFT_LINES + t;  // line = (bci, x, z)
  int z = (int)(l & 31);
  size_t r = l >> 5;
  int xx = (int)(r & 63);
  size_t bci = r >> 6;
  const cpx* src = in + ((bci * 64 + xx) * 64) * 32 + z;  // stride 32 along y
#pragma unroll
  for (int y = 0; y < 64; ++y)
    lds[y * FFT_LINES + t] = src[(size_t)y * 32];
  fft64_col(lds + t);
  cpx* dst = out + ((bci * 64 + xx) * 32) * 32 + z;
#pragma unroll
  for (int yk = 0; yk < 32; ++yk) {
    int bin = (yk < 16) ? yk : yk + 32;
    dst[(size_t)yk * 32] = lds[brev6(bin) * FFT_LINES + t];
  }
}

// ---------------- Forward X: 64 -> 16 kept bins (0..15)
__global__ __launch_bounds__(FFT_LINES) void k_fft_x_fwd(
    const cpx* __restrict__ in, cpx* __restrict__ out) {
  __shared__ cpx lds[64 * FFT_LINES];
  int t = threadIdx.x;
  size_t l = (size_t)blockIdx.x * FFT_LINES + t;  // line = (bci, yk, z)
  int z = (int)(l & 31);
  int yk = (int)((l >> 5) & 31);
  size_t bci = l >> 10;
  const cpx* src = in + ((bci * 64) * 32 + yk) * 32 + z;  // stride 1024 along x
#pragma unroll
  for (int xx = 0; xx < 64; ++xx)
    lds[xx * FFT_LINES + t] = src[(size_t)xx * 1024];
  fft64_col(lds + t);
  cpx* dst = out + ((bci * 16) * 32 + yk) * 32 + z;
#pragma unroll
  for (int xk = 0; xk < 16; ++xk)
    dst[(size_t)xk * 1024] = lds[brev6(xk) * FFT_LINES + t];
}

// ---------------- WMMA spectral contraction --------------------------------
// One wave per frequency point (xx, yk, zk); 4 freqs (consecutive zk) per
// 128-thread block. Complex GEMM as real GEMM:
//   A [16 x 64]: rows 0..7 = batches, rows 8..15 zero; k<32 -> Re x, k>=32 -> Im x
//   B [64 x 64]: col j=2o+p; built from (wr, wi, -wi) per frequency
//   D [16 x 64]: rows 0..7 -> out_ft[b][o].re/.im
typedef float v2f __attribute__((ext_vector_type(2)));
typedef float v8f __attribute__((ext_vector_type(8)));

__global__ __launch_bounds__(128) void k_spectral_contract(
    const cpx* __restrict__ xf,                     // [8][32][16][32][32]
    const float* __restrict__ w1, const float* __restrict__ w2,
    const float* __restrict__ w3, const float* __restrict__ w4,
    cpx* __restrict__ yf)                           // [8][32][16][32][32]
{
  __shared__ float As[4][64][16];   // [z][k][m] 16 KB (rows 8..15 zero)
  __shared__ cpx   Ws[4][32][32];   // [z][ci][co] (wr,wi) 32 KB
  int tid  = threadIdx.x;
  int wid  = tid >> 5;              // local z 0..3
  int lane = tid & 31;

  int blk = blockIdx.x;             // 16*32*8 = 4096 blocks
  int zg  = blk & 7;                // group of 4 z's
  int yk  = (blk >> 3) & 31;
  int xx  = blk >> 8;               // 0..15
  int zk0 = zg * 4;
  const float* w = (yk < 16) ? ((zk0 < 16) ? w1 : w2)
                             : ((zk0 < 16) ? w3 : w4);
  int my  = yk & 15;
  int mz0 = zk0 & 15;

  // zero-fill padding rows m=8..15 of A panel (disjoint from staging writes)
  for (int e = tid; e < 4 * 64 * 8; e += 128) {
    int m = 8 + (e & 7);
    int k = (e >> 3) & 63;
    int z = e >> 9;
    As[z][k][m] = 0.f;
  }
  // stage A panel: x_ft[b][ci][xx][yk][zk0+z], 1024 cpx, coalesced-ish
  for (int it = 0; it < 8; ++it) {
    int e  = tid + it * 128;
    int z  = e & 3;
    int ci = (e >> 2) & 31;
    int b  = e >> 7;
    cpx v = xf[(((size_t)(b * 32 + ci) * 16 + xx) * 32 + yk) * 32 + (zk0 + z)];
    As[z][ci][b]      = v.x;
    As[z][32 + ci][b] = v.y;
  }
  // stage W panel: per (ci,co) 4 mz * 2 floats = 32 B contiguous
  for (int it = 0; it < 8; ++it) {
    int pair = tid + it * 128;  // ci*32+co
    const float4* src = (const float4*)(
        w + ((((size_t)pair * 16 + xx) * 16 + my) * 16 + mz0) * 2);
    float4 q0 = src[0], q1 = src[1];
    int ci = pair >> 5, co = pair & 31;
    Ws[0][ci][co] = {q0.x, q0.y};
    Ws[1][ci][co] = {q0.z, q0.w};
    Ws[2][ci][co] = {q1.x, q1.y};
    Ws[3][ci][co] = {q1.z, q1.w};
  }
  __syncthreads();

  int z    = wid;
  int half = lane >> 4;     // A/B fragment half (K offset 0 or 2)
  int m    = lane & 15;     // A row / D column selector
  int n    = lane & 15;     // B column within tile

  // A fragments for all 16 K-steps: A.x=K(4s+2h), A.y=K(4s+2h+1), row m
  v2f a[16];
#pragma unroll
  for (int s = 0; s < 16; ++s) {
    int k0 = 4 * s + 2 * half;
    a[s].x = As[z][k0][m];
    a[s].y = As[z][k0 + 1][m];
  }

#pragma unroll
  for (int t = 0; t < 4; ++t) {               // N tiles (cols 16t..16t+15)
    v8f acc = {0.f, 0.f, 0.f, 0.f, 0.f, 0.f, 0.f, 0.f};
    int j = t * 16 + n;
    int o = j >> 1;
    int p = j & 1;
#pragma unroll
    for (int s = 0; s < 16; ++s) {            // K steps of 4
      int k0 = 4 * s + 2 * half;
      v2f bf;
      {
        int k = k0; int kk = k & 31;
        cpx wv = Ws[z][kk][o];
        bf.x = (k < 32) ? (p ? wv.y : wv.x) : (p ? wv.x : -wv.y);
      }
      {
        int k = k0 + 1; int kk = k & 31;
        cpx wv = Ws[z][kk][o];
        bf.y = (k < 32) ? (p ? wv.y : wv.x) : (p ? wv.x : -wv.y);
      }
      acc = __builtin_amdgcn_wmma_f32_16x16x4_f32(
          false, a[s], false, bf, (short)0, acc, false, false);
    }
    // D layout: lanes 0..15 hold M=r in VGPR r -> batch=r, col j
    if (lane < 16) {
#pragma unroll
      for (int r = 0; r < 8; ++r) {
        size_t idx = (((size_t)(r * 32 + o) * 16 + xx) * 32 + yk) * 32 + (zk0 + z);
        ((float*)yf)[idx * 2 + p] = acc[r];
      }
    }
  }
}

// ---------------- Inverse X: 16 bins -> 64 samples (x1/64)
__global__ __launch_bounds__(FFT_LINES) void k_fft_x_inv(
    const cpx* __restrict__ in, cpx* __restrict__ out) {
  __shared__ cpx lds[64 * FFT_LINES];
  const float S = 1.f / 64.f;
  int t = threadIdx.x;
  size_t l = (size_t)blockIdx.x * FFT_LINES + t;  // line = (bco, yk, z)
  int z = (int)(l & 31);
  int yk = (int)((l >> 5) & 31);
  size_t bco = l >> 10;
  const cpx* src = in + ((bco * 16) * 32 + yk) * 32 + z;
#pragma unroll
  for (int xk = 0; xk < 16; ++xk) {
    cpx v = src[(size_t)xk * 1024];
    lds[xk * FFT_LINES + t] = {v.x, -v.y};       // conj for inverse trick
  }
#pragma unroll
  for (int i = 16; i < 64; ++i)
    lds[i * FFT_LINES + t] = {0.f, 0.f};
  fft64_col(lds + t);
  cpx* dst = out + ((bco * 64) * 32 + yk) * 32 + z;
#pragma unroll
  for (int xx = 0; xx < 64; ++xx) {
    cpx v = lds[brev6(xx) * FFT_LINES + t];
    dst[(size_t)xx * 1024] = {v.x * S, -v.y * S};
  }
}

// ---------------- Inverse Y: bins {0..15 -> 0..15, 16..31 -> 48..63} (x1/64)
__global__ __launch_bounds__(FFT_LINES) void k_fft_y_inv(
    const cpx* __restrict__ in, cpx* __restrict__ out) {
  __shared__ cpx lds[64 * FFT_LINES];
  const float S = 1.f / 64.f;
  int t = threadIdx.x;
  size_t l = (size_t)blockIdx.x * FFT_LINES + t;  // line = (bco, xx, z)
  int z = (int)(l & 31);
  size_t r = l >> 5;
  int xx = (int)(r & 63);
  size_t bco = r >> 6;
  const cpx* src = in + ((bco * 64 + xx) * 32) * 32 + z;
#pragma unroll
  for (int yk = 0; yk < 16; ++yk) {
    cpx v = src[(size_t)yk * 32];
    lds[yk * FFT_LINES + t] = {v.x, -v.y};
  }
#pragma unroll
  for (int i = 16; i < 48; ++i)
    lds[i * FFT_LINES + t] = {0.f, 0.f};
#pragma unroll
  for (int yk = 16; yk < 32; ++yk) {
    cpx v = src[(size_t)yk * 32];
    lds[(yk + 32) * FFT_LINES + t] = {v.x, -v.y};
  }
  fft64_col(lds + t);
  cpx* dst = out + ((bco * 64 + xx) * 64) * 32 + z;
#pragma unroll
  for (int y = 0; y < 64; ++y) {
    cpx v = lds[brev6(y) * FFT_LINES + t];
    dst[(size_t)y * 32] = {v.x * S, -v.y * S};
  }
}

// ---------------- Inverse Z: Hermitian 32 kept bins -> 64 reals (x1/64)
__global__ __launch_bounds__(FFT_LINES) void k_fft_z_inv(
    const cpx* __restrict__ in, float* __restrict__ out) {
  __shared__ cpx lds[64 * FFT_LINES];
  const float S = 1.f / 64.f;
  int t = threadIdx.x;
  size_t l = (size_t)blockIdx.x * FFT_LINES + t;  // line = (bco, xx, y)
  const cpx* src = in + l * 32;
#pragma unroll
  for (int i = 0; i < 16; ++i) {
    cpx v = src[i];
    lds[i * FFT_LINES + t] = {v.x, -v.y};        // conj(spec[i])
  }
  lds[16 * FFT_LINES + t] = {0.f, 0.f};           // dropped bin 16
#pragma unroll
  for (int i = 17; i <= 32; ++i) {
    cpx v = src[i - 1];
    lds[i * FFT_LINES + t] = {v.x, -v.y};
  }
#pragma unroll
  for (int b2 = 33; b2 < 64; ++b2) {              // Hermitian mirror
    cpx c = lds[(64 - b2) * FFT_LINES + t];       // = conj(spec[64-b2])
    lds[b2 * FFT_LINES + t] = {c.x, -c.y};        // conj(full[b2]) = spec[64-b2]
  }
  fft64_col(lds + t);
  float4* dst = (float4*)(out + l * 64);
#pragma unroll
  for (int n = 0; n < 64; n += 4) {
    float4 v;
    v.x = lds[brev6(n + 0) * FFT_LINES + t].x * S;
    v.y = lds[brev6(n + 1) * FFT_LINES + t].x * S;
    v.z = lds[brev6(n + 2) * FFT_LINES + t].x * S;
    v.w = lds[brev6(n + 3) * FFT_LINES + t].x * S;
    dst[n >> 2] = v;
  }
}

// ---------------------------------------------------------------------------
extern "C" void kernel_launch(void* const* d_in, const int* in_sizes, int n_in,
                              void* d_out, int out_size, void* d_ws, size_t ws_size,
                              hipStream_t stream) {
  (void)in_sizes; (void)n_in; (void)out_size; (void)ws_size;
  const float* x  = (const float*)d_in[0];
  const float* w1 = (const float*)d_in[1];
  const float* w2 = (const float*)d_in[2];
  const float* w3 = (const float*)d_in[3];
  const float* w4 = (const float*)d_in[4];

  // Workspace layout (needs ~470 MB):
  cpx* buf1 = (cpx*)d_ws;                 // [8][32][64][64][32]  268 MB
  cpx* buf2 = buf1 + 33554432ull;         // [8][32][64][32][32]  134 MB
  cpx* buf3 = buf2 + 16777216ull;         // [8][32][16][32][32]   33.5 MB
  cpx* buf4 = buf3 + 4194304ull;          // [8][32][16][32][32]   33.5 MB

  k_fft_z_fwd<<<8192, FFT_LINES, 0, stream>>>(x, buf1);
  k_fft_y_fwd<<<4096, FFT_LINES, 0, stream>>>(buf1, buf2);
  k_fft_x_fwd<<<2048, FFT_LINES, 0, stream>>>(buf2, buf3);
  k_spectral_contract<<<4096, 128, 0, stream>>>(buf3, w1, w2, w3, w4, buf4);
  k_fft_x_inv<<<2048, FFT_LINES, 0, stream>>>(buf4, buf2);   // reuse buf2
  k_fft_y_inv<<<4096, FFT_LINES, 0, stream>>>(buf2, buf1);   // reuse buf1
  k_fft_z_inv<<<8192, FFT_LINES, 0, stream>>>(buf1, (float*)d_out);
}